// CRFLayer_22247930593676
// MI455X (gfx1250) — compile-verified
//
#include <hip/hip_runtime.h>
#include <math.h>

// ---------------------------------------------------------------------------
// CRF mean-field layer for MI455X (gfx1250), wave32 + WMMA f32 16x16x4.
//
// Both 21x21 kernels are separable Gaussians with a zeroed center:
//   conv2d(src,K) = colconv(rowconv(src)) - gn[10]^2 * src
// Each 1D 21-tap conv over a 16x16 tile is a GEMM with a Toeplitz weight
// matrix, executed as 9 chained v_wmma_f32_16x16x4_f32 (full f32 precision —
// var_I = mean_II - mean_I^2 cancels catastrophically, so no f16 inputs).
// Tiles + halo are staged in LDS; interior tiles fill with b128 copies
// (async-to-LDS via the gfx1250 builtin), edge tiles take a checked path.
// All planes live in d_ws (~47 MB, resident in the 192 MB L2).
// ---------------------------------------------------------------------------

typedef __attribute__((ext_vector_type(2))) float v2f;
typedef __attribute__((ext_vector_type(8))) float v8f;
typedef __attribute__((ext_vector_type(4))) int   v4i;

#define AS1 __attribute__((address_space(1)))
#define AS3 __attribute__((address_space(3)))

#if defined(__has_builtin)
#if __has_builtin(__builtin_amdgcn_global_load_async_to_lds_b128)
#define USE_ASYNC_LDS 1
#endif
#endif

#define IMG_H 512
#define IMG_W 512
#define NPIX  (IMG_H * IMG_W)
#define RAD   10
#define KTAPS 21
#define CRF_EPS 1e-4f

#define TAB_FLOATS 64            // ws[0..20]=gauss gn, ws[32..52]=gf gn, ws[60]=c_gauss, ws[61]=c_gf
#define PSIZE ((size_t)NPIX)

enum {
  P_GRAY  = 0,
  P_Q0    = 1,   // 4 planes: current Q
  P_MEANI = 5,
  P_VARI  = 6,
  P_SNORM = 7,
  P_BNORM = 8,
  P_A0    = 9,   // 4 planes
  P_B0    = 13,  // 4 planes
  P_SP0   = 17,  // 4 planes (spatial_out)
  P_ROW0  = 21,  // 12 planes: row-conv intermediates
  P_COL0  = 33,  // 12 planes: full (row+col) conv results
  N_PLANES = 45
};

#define DF_MULGRAY (1 << 16)     // multiply source by gray during load
#define DF_GF      (1 << 17)     // use guided-filter kernel (theta=10/3), else gauss (theta=2.5)
#define DF_ONES    (1 << 18)     // source is the constant-1 field

struct ConvDesc { int d[12]; };

__device__ __forceinline__ float* plane(float* ws, int p) {
  return ws + TAB_FLOATS + (size_t)p * PSIZE;
}

// ---------------------------------------------------------------------------
// 1) Normalized 1D 21-tap tables + center-correction constants.
// ---------------------------------------------------------------------------
__global__ void k_init_tab(float* ws) {
  int t = threadIdx.x;
  if (t < 2 * KTAPS) {
    int set = t / KTAPS;                       // 0: gauss (theta=2.5), 1: gf (theta=10/3)
    int i   = t % KTAPS;
    float theta  = (set == 0) ? 2.5f : (10.0f / 3.0f);
    float inv2t2 = 1.0f / (2.0f * theta * theta);
    float s = 0.0f;
    for (int j = 0; j < KTAPS; ++j) { float d = (float)(j - RAD); s += expf(-d * d * inv2t2); }
    float d  = (float)(i - RAD);
    float gi = expf(-d * d * inv2t2) / s;
    ws[set * 32 + i] = gi;
    if (i == RAD) ws[60 + set] = gi * gi;      // ws[60]=c_gauss, ws[61]=c_gf
  }
}

// ---------------------------------------------------------------------------
// 2) gray = image . [0.2989,0.5870,0.1140];  Q0 = softmax(-unary)
// ---------------------------------------------------------------------------
__global__ void k_prep(const float* __restrict__ unary, const float* __restrict__ image,
                       float* ws) {
  int i = blockIdx.x * blockDim.x + threadIdx.x;
  if (i >= NPIX) return;
  float r = image[i * 3 + 0], g = image[i * 3 + 1], b = image[i * 3 + 2];
  plane(ws, P_GRAY)[i] = 0.2989f * r + 0.5870f * g + 0.1140f * b;

  float l0 = -unary[i * 4 + 0], l1 = -unary[i * 4 + 1];
  float l2 = -unary[i * 4 + 2], l3 = -unary[i * 4 + 3];
  float mx = fmaxf(fmaxf(l0, l1), fmaxf(l2, l3));
  float e0 = expf(l0 - mx), e1 = expf(l1 - mx), e2 = expf(l2 - mx), e3 = expf(l3 - mx);
  float inv = 1.0f / (e0 + e1 + e2 + e3);
  plane(ws, P_Q0 + 0)[i] = e0 * inv;
  plane(ws, P_Q0 + 1)[i] = e1 * inv;
  plane(ws, P_Q0 + 2)[i] = e2 * inv;
  plane(ws, P_Q0 + 3)[i] = e3 * inv;
}

// ---------------------------------------------------------------------------
// Zero-padded weight table in LDS: swpad[16+i] = w[i], rest 0, so the
// Toeplitz index i0 = 4kk+k0-m (range [-15,38]) maps to [1,54] -> no bounds
// checks in the A-fragment gather.
// ---------------------------------------------------------------------------
__device__ __forceinline__ void load_wpad(float* swpad, const float* ws, int wset) {
  int t = threadIdx.x;
  if (t < 64) {
    float v = 0.0f;
    if (t >= 16 && t < 16 + KTAPS) v = ws[wset * 32 + (t - 16)];
    swpad[t] = v;
  }
}

// ---------------------------------------------------------------------------
// 3) Row conv: out[y, x0+m] = sum_t w[t]*in[y, x0+m+t-RAD].
//    One 16x16 tile per wave. Tile+halo = 16 rows x 48 cols (origin x0-16),
//    LDS stride 52: rows 16B-aligned for b128 fills AND conflict-free column
//    reads (52=4*13; 13*m mod 16 is a permutation; half-waves differ mod 4).
// ---------------------------------------------------------------------------
#define RT_W 48
#define RT_S 52

__global__ void __launch_bounds__(256) k_rowconv(float* ws, ConvDesc desc, int outBase) {
  __shared__ float swpad[64];
  __shared__ float tile[8 * 16 * RT_S];

  const int z  = blockIdx.z;
  const int dc = desc.d[z];
  load_wpad(swpad, ws, (dc & DF_GF) ? 1 : 0);

  const int lane = threadIdx.x & 31;
  const int wave = threadIdx.x >> 5;
  const int x0 = (blockIdx.x * 8 + wave) * 16;
  const int y0 = blockIdx.y * 16;

  const bool ones = (dc & DF_ONES) != 0;
  const bool mulg = (dc & DF_MULGRAY) != 0;
  const float* src = ones ? nullptr : plane(ws, dc & 0xFFFF);
  const float* gry = plane(ws, P_GRAY);

  float* tl = tile + wave * (16 * RT_S);
  const bool xedge = (x0 < 16) || (x0 + 32 > IMG_W);

  if (!xedge) {
    // interior: all 48 halo columns in-bounds -> vector fill, 6 b128/lane
#if USE_ASYNC_LDS
    if (!ones && !mulg) {
#pragma unroll
      for (int j = 0; j < 6; ++j) {
        int fidx = j * 32 + lane;              // 0..191 float4s
        int r = fidx / 12, c4 = fidx % 12;
        const float* gp = src + (size_t)(y0 + r) * IMG_W + (x0 - 16) + 4 * c4;
        float* lp = tl + r * RT_S + 4 * c4;
        __builtin_amdgcn_global_load_async_to_lds_b128((AS1 v4i*)gp, (AS3 v4i*)lp, 0, 0);
      }
      asm volatile("s_wait_asynccnt 0x0" ::: "memory");
    } else
#endif
    {
#pragma unroll
      for (int j = 0; j < 6; ++j) {
        int fidx = j * 32 + lane;
        int r = fidx / 12, c4 = fidx % 12;
        size_t gi = (size_t)(y0 + r) * IMG_W + (x0 - 16) + 4 * c4;
        float4 v;
        if (ones) v = make_float4(1.f, 1.f, 1.f, 1.f);
        else      v = *(const float4*)&src[gi];
        if (mulg) {
          float4 gv = *(const float4*)&gry[gi];
          v.x *= gv.x; v.y *= gv.y; v.z *= gv.z; v.w *= gv.w;
        }
        *(float4*)&tl[r * RT_S + 4 * c4] = v;
      }
    }
  } else {
    // x-edge tiles (2 of 32): per-element with bounds checks
#pragma unroll
    for (int j = 0; j < 24; ++j) {
      int idx = j * 32 + lane;                 // 0..767
      int r   = idx / RT_W;
      int c   = idx - r * RT_W;
      int gx  = x0 - 16 + c;
      float v = 0.0f;
      if (gx >= 0 && gx < IMG_W) {
        size_t gi = (size_t)(y0 + r) * IMG_W + gx;
        v = ones ? 1.0f : src[gi];
        if (mulg) v *= gry[gi];
      }
      tl[r * RT_S + c] = v;
    }
  }
  __syncthreads();

  // ---- branch-free A fragments + WMMA chain ----
  const int m  = lane & 15;
  const int k0 = (lane >> 4) * 2;
  v2f af[9];
#pragma unroll
  for (int kk = 0; kk < 9; ++kk) {
    const int p = 4 * kk + k0 - m + 16;        // in [1,54]
    af[kk].x = swpad[p];
    af[kk].y = swpad[p + 1];
  }

  const float* brow = tl + m * RT_S;           // N-lane's image row
  v8f acc = {0.f, 0.f, 0.f, 0.f, 0.f, 0.f, 0.f, 0.f};
#pragma unroll
  for (int kk = 0; kk < 9; ++kk) {
    const int c = 4 * kk + k0 + 6;             // (4kk+k0-10) - (x0-16), in [6,42]
    v2f bf; bf.x = brow[c]; bf.y = brow[c + 1];
    acc = __builtin_amdgcn_wmma_f32_16x16x4_f32(false, af[kk], false, bf,
                                                (short)0, acc, false, false);
  }

  // D: VGPR j -> x = x0 + j + 8*(lane>=16), row = y0 + (lane&15).
  float* out = plane(ws, outBase + z) + (size_t)(y0 + m) * IMG_W + x0 + 8 * (lane >> 4);
  *(float4*)(out)     = make_float4(acc[0], acc[1], acc[2], acc[3]);
  *(float4*)(out + 4) = make_float4(acc[4], acc[5], acc[6], acc[7]);
}

// ---------------------------------------------------------------------------
// 4) Col conv: same GEMM with N = columns, M = output y. Tile+halo is
//    48 rows x 16 cols (origin y0-16), stride 16: half-waves read rows k and
//    k+2 -> bank sets offset by 32 -> conflict-free. Rows are all-or-nothing
//    in bounds, so the vector fill needs no partial handling.
// ---------------------------------------------------------------------------
__global__ void __launch_bounds__(256) k_colconv(float* ws, ConvDesc desc, int inBase, int outBase) {
  __shared__ float swpad[64];
  __shared__ float tile[8 * 48 * 16];

  const int z  = blockIdx.z;
  const int dc = desc.d[z];
  load_wpad(swpad, ws, (dc & DF_GF) ? 1 : 0);

  const int lane = threadIdx.x & 31;
  const int wave = threadIdx.x >> 5;
  const int x0 = (blockIdx.x * 8 + wave) * 16;
  const int y0 = blockIdx.y * 16;

  const float* src = plane(ws, inBase + z);
  float* tl = tile + wave * (48 * 16);
  const bool yedge = (y0 < 16) || (y0 + 32 > IMG_H);

#if USE_ASYNC_LDS
  if (!yedge) {
#pragma unroll
    for (int j = 0; j < 6; ++j) {
      int fidx = j * 32 + lane;                // 0..191 float4s
      int r = fidx >> 2, c4 = fidx & 3;        // r in 0..47
      const float* gp = src + (size_t)(y0 - 16 + r) * IMG_W + x0 + 4 * c4;
      float* lp = tl + r * 16 + 4 * c4;
      __builtin_amdgcn_global_load_async_to_lds_b128((AS1 v4i*)gp, (AS3 v4i*)lp, 0, 0);
    }
    asm volatile("s_wait_asynccnt 0x0" ::: "memory");
  } else
#endif
  {
#pragma unroll
    for (int j = 0; j < 6; ++j) {
      int fidx = j * 32 + lane;
      int r = fidx >> 2, c4 = fidx & 3;
      int gy = y0 - 16 + r;
      float4 v = make_float4(0.f, 0.f, 0.f, 0.f);
      if (gy >= 0 && gy < IMG_H)
        v = *(const float4*)&src[(size_t)gy * IMG_W + x0 + 4 * c4];
      *(float4*)&tl[r * 16 + 4 * c4] = v;
    }
  }
  __syncthreads();

  const int m  = lane & 15;
  const int k0 = (lane >> 4) * 2;
  v2f af[9];
#pragma unroll
  for (int kk = 0; kk < 9; ++kk) {
    const int p = 4 * kk + k0 - m + 16;
    af[kk].x = swpad[p];
    af[kk].y = swpad[p + 1];
  }

  v8f acc = {0.f, 0.f, 0.f, 0.f, 0.f, 0.f, 0.f, 0.f};
#pragma unroll
  for (int kk = 0; kk < 9; ++kk) {
    const int rb = 4 * kk + k0 + 6;            // tile row for K index, in [6,42]
    v2f bf; bf.x = tl[rb * 16 + m]; bf.y = tl[(rb + 1) * 16 + m];
    acc = __builtin_amdgcn_wmma_f32_16x16x4_f32(false, af[kk], false, bf,
                                                (short)0, acc, false, false);
  }

  float* out = plane(ws, outBase + z);
  const int x = x0 + m;
  const int ybase = y0 + 8 * (lane >> 4);
#pragma unroll
  for (int j = 0; j < 8; ++j) out[(size_t)(ybase + j) * IMG_W + x] = acc[j];
}

// ---------------------------------------------------------------------------
// 5) Pointwise: iteration-invariant stats (p = ones case of guided filter).
// ---------------------------------------------------------------------------
__global__ void k_pw_setup(float* ws) {
  int i = blockIdx.x * blockDim.x + threadIdx.x;
  if (i >= NPIX) return;
  float cga = ws[60], cgf = ws[61];
  float g   = plane(ws, P_GRAY)[i];
  float Nf  = plane(ws, P_COL0 + 0)[i] - cgf;
  float sn  = plane(ws, P_COL0 + 1)[i] - cga;
  float mI  = plane(ws, P_COL0 + 2)[i] - cgf * g;
  float mII = plane(ws, P_COL0 + 3)[i] - cgf * g * g;
  float var = mII - mI * mI;
  float a0  = mI * (1.0f - Nf) / (var + CRF_EPS);    // cov_Ip = mI - mI*N for p=1
  float b0  = Nf - a0 * mI;
  plane(ws, P_MEANI)[i] = mI;
  plane(ws, P_VARI)[i]  = var;
  plane(ws, P_SNORM)[i] = sn;
  plane(ws, P_A0)[i] = a0;
  plane(ws, P_B0)[i] = b0;
}

__global__ void k_pw_bnorm(float* ws) {
  int i = blockIdx.x * blockDim.x + threadIdx.x;
  if (i >= NPIX) return;
  float cgf = ws[61];
  float g   = plane(ws, P_GRAY)[i];
  float gfa = plane(ws, P_COL0 + 0)[i] - cgf * plane(ws, P_A0)[i];
  float gfb = plane(ws, P_COL0 + 1)[i] - cgf * plane(ws, P_B0)[i];
  plane(ws, P_BNORM)[i] = gfa * g + gfb;
}

// ---------------------------------------------------------------------------
// 6) Per-iteration stage 1: a_c, b_c, spatial_out from COL[0..11].
// ---------------------------------------------------------------------------
__global__ void k_pw_iter1(float* ws) {
  int i = blockIdx.x * blockDim.x + threadIdx.x;
  if (i >= NPIX) return;
  float cga = ws[60], cgf = ws[61];
  float g   = plane(ws, P_GRAY)[i];
  float mI  = plane(ws, P_MEANI)[i];
  float var = plane(ws, P_VARI)[i];
  float sn  = plane(ws, P_SNORM)[i];
  float inv_var = 1.0f / (var + CRF_EPS);
  float inv_sn  = 1.0f / sn;
#pragma unroll
  for (int c = 0; c < 4; ++c) {
    float q   = plane(ws, P_Q0 + c)[i];
    float mp  = plane(ws, P_COL0 + c)[i]     - cgf * q;          // mean_p
    float mIp = plane(ws, P_COL0 + 4 + c)[i] - cgf * g * q;      // mean_Ip
    float gq  = plane(ws, P_COL0 + 8 + c)[i] - cga * q;          // gaussian(Q)
    float a   = (mIp - mI * mp) * inv_var;
    float b   = mp - a * mI;
    plane(ws, P_A0 + c)[i]  = a;
    plane(ws, P_B0 + c)[i]  = b;
    plane(ws, P_SP0 + c)[i] = gq * inv_sn;                       // spatial_out
  }
}

// ---------------------------------------------------------------------------
// 7) Per-iteration stage 2: bilateral_out + softmax update of Q.
// ---------------------------------------------------------------------------
__global__ void k_pw_iter2(float* ws, const float* __restrict__ unary,
                           float* __restrict__ out, int writeOut) {
  int i = blockIdx.x * blockDim.x + threadIdx.x;
  if (i >= NPIX) return;
  float cgf = ws[61];
  float g   = plane(ws, P_GRAY)[i];
  float inv_bn = 1.0f / plane(ws, P_BNORM)[i];
  float lg[4];
  float mx = -3.4e38f;
#pragma unroll
  for (int c = 0; c < 4; ++c) {
    float gfa = plane(ws, P_COL0 + c)[i]     - cgf * plane(ws, P_A0 + c)[i];
    float gfb = plane(ws, P_COL0 + 4 + c)[i] - cgf * plane(ws, P_B0 + c)[i];
    float bil = (gfa * g + gfb) * inv_bn;
    float msg = 3.0f * plane(ws, P_SP0 + c)[i] + 10.0f * bil;
    lg[c] = -unary[i * 4 + c] + msg;
    mx = fmaxf(mx, lg[c]);
  }
  float s = 0.f;
#pragma unroll
  for (int c = 0; c < 4; ++c) { lg[c] = expf(lg[c] - mx); s += lg[c]; }
  float inv = 1.0f / s;
#pragma unroll
  for (int c = 0; c < 4; ++c) {
    float q = lg[c] * inv;
    plane(ws, P_Q0 + c)[i] = q;
    if (writeOut) out[i * 4 + c] = q;
  }
}

// ---------------------------------------------------------------------------
// Orchestration: ~37 launches; global sync between conv passes is required
// (receptive field grows ~50 px/iteration -> no cross-iteration fusion).
// ---------------------------------------------------------------------------
extern "C" void kernel_launch(void* const* d_in, const int* in_sizes, int n_in,
                              void* d_out, int out_size, void* d_ws, size_t ws_size,
                              hipStream_t stream) {
  const float* unary = (const float*)d_in[0];   // [1,512,512,4] f32
  const float* image = (const float*)d_in[1];   // [1,512,512,3] f32
  float* out = (float*)d_out;                   // [1,512,512,4] f32
  float* ws  = (float*)d_ws;                    // needs >= 64 + 45*NPIX floats (~47 MB)

  const int PW_BLOCKS = NPIX / 256;

  k_init_tab<<<1, 64, 0, stream>>>(ws);
  k_prep<<<PW_BLOCKS, 256, 0, stream>>>(unary, image, ws);

  // --- setup: gf(1), gauss(1), gf(gray), gf(gray^2) ---
  ConvDesc dsS = {};
  dsS.d[0] = DF_ONES | DF_GF;
  dsS.d[1] = DF_ONES;                       // gauss
  dsS.d[2] = P_GRAY | DF_GF;
  dsS.d[3] = P_GRAY | DF_GF | DF_MULGRAY;   // gray^2
  dim3 g4(4, 32, 4);
  k_rowconv<<<g4, 256, 0, stream>>>(ws, dsS, P_ROW0);
  k_colconv<<<g4, 256, 0, stream>>>(ws, dsS, P_ROW0, P_COL0);
  k_pw_setup<<<PW_BLOCKS, 256, 0, stream>>>(ws);

  // --- setup: gf(a0), gf(b0) -> bilateral_norm ---
  ConvDesc dsN = {};
  dsN.d[0] = P_A0 | DF_GF;
  dsN.d[1] = P_B0 | DF_GF;
  dim3 g2(4, 32, 2);
  k_rowconv<<<g2, 256, 0, stream>>>(ws, dsN, P_ROW0);
  k_colconv<<<g2, 256, 0, stream>>>(ws, dsN, P_ROW0, P_COL0);
  k_pw_bnorm<<<PW_BLOCKS, 256, 0, stream>>>(ws);

  // --- 5 mean-field iterations ---
  for (int it = 0; it < 5; ++it) {
    ConvDesc d12 = {};
    for (int c = 0; c < 4; ++c) {
      d12.d[c]     = (P_Q0 + c) | DF_GF;                // gf(Q)
      d12.d[4 + c] = (P_Q0 + c) | DF_GF | DF_MULGRAY;   // gf(gray*Q)
      d12.d[8 + c] = (P_Q0 + c);                        // gauss(Q)
    }
    dim3 g12(4, 32, 12);
    k_rowconv<<<g12, 256, 0, stream>>>(ws, d12, P_ROW0);
    k_colconv<<<g12, 256, 0, stream>>>(ws, d12, P_ROW0, P_COL0);
    k_pw_iter1<<<PW_BLOCKS, 256, 0, stream>>>(ws);

    ConvDesc d8 = {};
    for (int c = 0; c < 4; ++c) {
      d8.d[c]     = (P_A0 + c) | DF_GF;                 // gf(a)
      d8.d[4 + c] = (P_B0 + c) | DF_GF;                 // gf(b)
    }
    dim3 g8(4, 32, 8);
    k_rowconv<<<g8, 256, 0, stream>>>(ws, d8, P_ROW0);
    k_colconv<<<g8, 256, 0, stream>>>(ws, d8, P_ROW0, P_COL0);
    k_pw_iter2<<<PW_BLOCKS, 256, 0, stream>>>(ws, unary, out, (it == 4) ? 1 : 0);
  }
}